// entrop_83880711291387
// MI455X (gfx1250) — compile-verified
//
#include <hip/hip_runtime.h>
#include <hip/hip_bf16.h>

// ---- problem constants (match reference) ------------------------------------
#define BATCH    16
#define CHAN     3
#define HDIM     1024
#define WDIM     1024
#define N_TOTAL  (BATCH*CHAN*HDIM*WDIM)      // 50,331,648 elements
#define NQ       (N_TOTAL/4)                 // float4 count
#define N_PATCHES 64                         // 8x8 spatial patches
#define PATCH_ELEMS 786432                   // 16*3*128*128 elements per patch
#define HIST_WORDS (N_PATCHES*256)

typedef __attribute__((ext_vector_type(2))) float v2f;
typedef __attribute__((ext_vector_type(4))) float v4f;
typedef __attribute__((ext_vector_type(8))) float v8f;

// ---- kernel 1: zero histogram workspace + output ----------------------------
__global__ void ew_init_kernel(unsigned int* __restrict__ hist, float* __restrict__ out) {
    int i = blockIdx.x * blockDim.x + threadIdx.x;
    if (i < HIST_WORDS) hist[i] = 0u;
    if (i == 0) out[0] = 0.0f;
}

// ---- kernel 2: per-patch 256-bin histograms of hr ---------------------------
// grid = 64 patches * 8 parts; each block handles 6 of the 48 (b,c) slices of
// one spatial patch. Per-wave private 256-bin LDS histograms (8 KB of the
// 320 KB WGP LDS) so ds_add conflicts are confined to one wave32; single
// 8-way merge + one global atomic per bin at the end. hr loads use default RT
// temporal policy so the tensor fills the 192 MB L2 for reuse by the loss pass.
__global__ __launch_bounds__(256) void ew_hist_kernel(const float* __restrict__ hr,
                                                      unsigned int* __restrict__ hist) {
    __shared__ unsigned int lh[8 * 256];     // one 256-bin histogram per wave
    const int tid  = threadIdx.x;
    const int wave = tid >> 5;               // wave32 id, 0..7
#pragma unroll
    for (int k = 0; k < 8; ++k) lh[k * 256 + tid] = 0u;
    __syncthreads();

    unsigned int* __restrict__ myh = &lh[wave * 256];

    const int p    = blockIdx.x >> 3;     // patch id 0..63
    const int part = blockIdx.x & 7;      // slice group 0..7
    const int pi   = p >> 3;              // patch row
    const int pj   = p & 7;               // patch col
    const int sliceBase = part * 6;       // 6 slices of 48

    // 6 slices * 128 rows * 32 float4 per row = 24576 float4 loads per block
    for (int idx = tid; idx < 6 * 128 * 32; idx += 256) {
        const int quad = idx & 31;        // float4 within the 128-wide patch row
        const int rowg = idx >> 5;        // 0..767
        const int r    = rowg & 127;      // row within patch
        const int s    = rowg >> 7;       // slice within group (0..5)
        const long off = ((long)(sliceBase + s) * HDIM + (long)(pi * 128 + r)) * (long)WDIM
                         + (long)(pj * 128 + quad * 4);
        const float4 v = *(const float4*)(hr + off);
        int b0 = (int)(v.x * 255.0f); b0 = b0 < 0 ? 0 : (b0 > 255 ? 255 : b0);
        int b1 = (int)(v.y * 255.0f); b1 = b1 < 0 ? 0 : (b1 > 255 ? 255 : b1);
        int b2 = (int)(v.z * 255.0f); b2 = b2 < 0 ? 0 : (b2 > 255 ? 255 : b2);
        int b3 = (int)(v.w * 255.0f); b3 = b3 < 0 ? 0 : (b3 > 255 ? 255 : b3);
        atomicAdd(&myh[b0], 1u);
        atomicAdd(&myh[b1], 1u);
        atomicAdd(&myh[b2], 1u);
        atomicAdd(&myh[b3], 1u);
    }
    __syncthreads();

    unsigned int total = 0u;
#pragma unroll
    for (int k = 0; k < 8; ++k) total += lh[k * 256 + tid];
    atomicAdd(&hist[p * 256 + tid], total);
}

// ---- kernel 3: entropy per patch, normalize (e - min)/max -------------------
__global__ __launch_bounds__(64) void ew_entropy_kernel(const unsigned int* __restrict__ hist,
                                                        float* __restrict__ w64) {
    __shared__ float ent[N_PATCHES];
    __shared__ float red[2]; // {min, max}
    const int t = threadIdx.x; // patch id
    float e = 0.0f;
    const float invN = 1.0f / (float)PATCH_ELEMS;
    for (int b = 0; b < 256; ++b) {
        const unsigned int c = hist[t * 256 + b];
        if (c) {
            const float pr = (float)c * invN;
            e -= pr * log2f(pr);
        }
    }
    ent[t] = e;
    __syncthreads();
    if (t == 0) {
        float mn = ent[0], mx = ent[0];
        for (int i = 1; i < N_PATCHES; ++i) {
            mn = fminf(mn, ent[i]);
            mx = fmaxf(mx, ent[i]);
        }
        red[0] = mn;
        red[1] = mx; // original max, matching reference evaluation order
    }
    __syncthreads();
    w64[t] = (ent[t] - red[0]) / red[1];
}

// ---- kernel 4: sum(w * |sr - hr|) / N ---------------------------------------
// Grid-stride float4 streaming with NON-TEMPORAL loads: sr is read exactly
// once (don't let it claim L2), and this is hr's last use — so pass-2's
// RT-cached hr survives in the 192 MB L2 and the second hr read mostly hits,
// cutting HBM traffic from ~603 MB to ~410 MB (~18 us at 23.3 TB/s).
// Block reduction finishes with a wave32 V_WMMA_F32_16X16X4_F32: A holds
// per-lane partials (16x4 layout), B = ones, so D row m = v_m + v_{m+16};
// each lane sums its 8 D VGPRs, leaving the block total split across lanes
// 0 and 16.
__global__ __launch_bounds__(256) void ew_loss_kernel(const float* __restrict__ sr,
                                                      const float* __restrict__ hr,
                                                      const float* __restrict__ w64,
                                                      float* __restrict__ out) {
    __shared__ float wpatch[N_PATCHES];
    __shared__ float red[256];
    const int tid = threadIdx.x;
    if (tid < N_PATCHES) wpatch[tid] = w64[tid];
    __syncthreads();

    const v4f* __restrict__ sr4 = (const v4f*)sr;
    const v4f* __restrict__ hr4 = (const v4f*)hr;
    const int stride = (int)(gridDim.x * blockDim.x);

    float acc = 0.0f;
    for (int q = blockIdx.x * blockDim.x + tid; q < NQ; q += stride) {
        const int n   = q * 4;
        const int col = n & (WDIM - 1);
        const int h   = (n >> 10) & (HDIM - 1);
        const float w = wpatch[((h >> 7) << 3) | (col >> 7)];
        const v4f a = __builtin_nontemporal_load(sr4 + q);  // th:NT
        const v4f b = __builtin_nontemporal_load(hr4 + q);  // th:NT (last use)
        acc += w * (fabsf(a.x - b.x) + fabsf(a.y - b.y) +
                    fabsf(a.z - b.z) + fabsf(a.w - b.w));
    }
    red[tid] = acc;
    __syncthreads();

    if (tid < 32) {  // exactly wave 0 under wave32; EXEC all-ones inside
        float v = 0.0f;
#pragma unroll
        for (int k = 0; k < 8; ++k) v += red[tid + 32 * k];
        // A (16x4 f32): lane L -> M=L%16, VGPR0=K(0 or 2); set {v, 0}
        v2f a; a.x = v; a.y = 0.0f;
        v2f bm; bm.x = 1.0f; bm.y = 1.0f;   // B = ones(4x16)
        v8f c = {};
        v8f d = __builtin_amdgcn_wmma_f32_16x16x4_f32(
            /*neg_a=*/false, a, /*neg_b=*/false, bm,
            /*c_mod=*/(short)0, c, /*reuse_a=*/false, /*reuse_b=*/false);
        // lanes 0-15 all hold sum(v_0..7,v_16..23); lanes 16-31 the rest
        const float half = d[0] + d[1] + d[2] + d[3] + d[4] + d[5] + d[6] + d[7];
        if (tid == 0 || tid == 16) red[tid] = half;
    }
    __syncthreads();
    if (tid == 0) {
        atomicAdd(out, (red[0] + red[16]) * (1.0f / (float)N_TOTAL));
    }
}

// ---- launcher ---------------------------------------------------------------
extern "C" void kernel_launch(void* const* d_in, const int* in_sizes, int n_in,
                              void* d_out, int out_size, void* d_ws, size_t ws_size,
                              hipStream_t stream) {
    const float* sr = (const float*)d_in[0];
    const float* hr = (const float*)d_in[1];
    float* out = (float*)d_out;

    unsigned int* hist = (unsigned int*)d_ws;                 // 64*256 u32 = 64 KB
    float* w64 = (float*)((char*)d_ws + HIST_WORDS * sizeof(unsigned int)); // 64 f32

    ew_init_kernel<<<(HIST_WORDS + 255) / 256, 256, 0, stream>>>(hist, out);
    ew_hist_kernel<<<N_PATCHES * 8, 256, 0, stream>>>(hr, hist);
    ew_entropy_kernel<<<1, 64, 0, stream>>>(hist, w64);
    ew_loss_kernel<<<2048, 256, 0, stream>>>(sr, hr, w64, out);
}